// HierarchicalPositiveInverter_88905823027808
// MI455X (gfx1250) — compile-verified
//
#include <hip/hip_runtime.h>

typedef __attribute__((ext_vector_type(2))) float v2f;
typedef __attribute__((ext_vector_type(8))) float v8f;

#define B_ROWS 16384
#define DHID   1024
#define KB     32          // K-chunk staged per double-buffer step
#define A_STR  36          // padded LDS row stride for A (floats)
#define B_STR  68          // padded LDS row stride for B (floats)

// ---------------------------------------------------------------------------
// gfx1250 async global->LDS staging (ASYNCcnt-tracked). Inline asm per
// CDNA5 ISA / CDNA5 ISA (portable across toolchains).
// Each lane supplies its own LDS byte address (VDST) and 64-bit global
// address (VADDR); transfers 16B per lane.
// ---------------------------------------------------------------------------
__device__ __forceinline__ void async_ld_b128(unsigned lds_byte_off, const float* g)
{
    asm volatile("global_load_async_to_lds_b128 %0, %1, off"
                 :: "v"(lds_byte_off), "v"((unsigned long long)(uintptr_t)g)
                 : "memory");
}
__device__ __forceinline__ void wait_async0()
{
    asm volatile("s_wait_asynccnt 0x0" ::: "memory");
}
__device__ __forceinline__ unsigned lds_off(const void* p)
{
    // LDS aperture: hardware uses addr[31:0] (ISA 10.2 per-aperture calc)
    return (unsigned)(uintptr_t)p;
}

// ---------------------------------------------------------------------------
// Generic GEMM: Y[row, n] = act( sum_k X[row,k] * W[k,n] + bias[n] )
// Optional gather/scatter through perm[off[e]..off[e]+cnt[e]) for hard-routed
// expert dispatch. Block = 128 threads (4 wave32s); block tile = 16 x 64;
// wave tile = 16x16 via V_WMMA_F32_16X16X4_F32. A and B chunks are staged
// into LDS with async b128 copies, double buffered.
// ---------------------------------------------------------------------------
__global__ __launch_bounds__(128) void gemm_relu_wmma(
    const float* __restrict__ X, const float* __restrict__ W,
    const float* __restrict__ bias, float* __restrict__ Y,
    const int* __restrict__ perm, const int* __restrict__ offArr,
    const int* __restrict__ cntArr, int e, int N, int K, int doRelu)
{
    __shared__ float ldsA[2][16 * A_STR];
    __shared__ float ldsB[2][KB * B_STR];

    const int tid    = threadIdx.x;
    const int lane   = tid & 31;
    const int wave   = tid >> 5;
    const int rowBase = blockIdx.y * 16;

    const int cnt = cntArr ? cntArr[e] : (int)(gridDim.y * 16);
    if (rowBase >= cnt) return;
    const int off = offArr ? offArr[e] : 0;

    // ---- staging assignment -------------------------------------------------
    // A: 16 rows x KB floats = 512 floats -> 1 b128 per thread per chunk
    const int ar  = tid >> 3;                 // 0..15 : A row within tile
    const int seg = (tid & 7) * 4;            // float segment within chunk
    int rA  = rowBase + ar;
    int rAc = (rA < cnt) ? rA : (cnt - 1);
    const int srcRowA = perm ? perm[off + rAc] : rAc;
    const float* aSeg = X + (size_t)srcRowA * K + seg;

    // B: KB x 64 floats = 2048 floats -> 4 b128 per thread per chunk
    const int colBase = blockIdx.x * 64;

    // ---- compute-fragment coordinates --------------------------------------
    const int mLane  = lane & 15;
    const int kbHalf = (lane >> 4) * 2;       // lane-half selects K {0,1}/{2,3}
    const int n      = colBase + wave * 16 + mLane;

    // ---- staging helper -----------------------------------------------------
    auto stage = [&](int buf, int k0) {
        async_ld_b128(lds_off(&ldsA[buf][ar * A_STR + seg]), aSeg + k0);
#pragma unroll
        for (int i = 0; i < 4; ++i) {
            const int li = i * 512 + tid * 4;   // linear float index in chunk
            const int br = li >> 6;             // 0..31 : B row in chunk
            const int bc = li & 63;             // col within 64, multiple of 4
            async_ld_b128(lds_off(&ldsB[buf][br * B_STR + bc]),
                          W + (size_t)(k0 + br) * N + colBase + bc);
        }
    };

    v8f c = {};

    stage(0, 0);
    wait_async0();
    __syncthreads();

    int buf = 0;
    for (int k0 = 0; k0 < K; k0 += KB) {
        if (k0 + KB < K) stage(buf ^ 1, k0 + KB);

        const float* As = &ldsA[buf][mLane * A_STR + kbHalf];
        const float* Bs = &ldsB[buf][kbHalf * B_STR + wave * 16 + mLane];
#pragma unroll
        for (int kk = 0; kk < KB; kk += 4) {
            v2f a = *(const v2f*)(As + kk);
            v2f b;
            b.x = Bs[(size_t)kk * B_STR];
            b.y = Bs[(size_t)(kk + 1) * B_STR];
            c = __builtin_amdgcn_wmma_f32_16x16x4_f32(
                    /*neg_a=*/false, a, /*neg_b=*/false, b,
                    /*c_mod=*/(short)0, c, /*reuse_a=*/false, /*reuse_b=*/false);
        }

        wait_async0();
        __syncthreads();
        buf ^= 1;
    }

    // ---- epilogue: bias + relu, masked scatter ------------------------------
    const float bn   = bias[n];
    const int   mOff = (lane >> 4) * 8;       // C layout: lanes16-31 -> M+8
#pragma unroll
    for (int v = 0; v < 8; ++v) {
        const int rr = rowBase + v + mOff;
        if (rr < cnt) {
            const int dstRow = perm ? perm[off + rr] : rr;
            float val = c[v] + bn;
            if (doRelu) val = val > 0.f ? val : 0.f;
            Y[(size_t)dstRow * N + n] = val;
        }
    }
}

// ---------------------------------------------------------------------------
// Tiny second-layer heads (N2 <= 7): wave-per-row GEMV + shuffle reduction.
// ---------------------------------------------------------------------------
__global__ __launch_bounds__(128) void head2_gemv(
    const float* __restrict__ H, const float* __restrict__ W2,
    const float* __restrict__ b2, float* __restrict__ outS, int N2, int K)
{
    const int lane = threadIdx.x & 31;
    const int wave = threadIdx.x >> 5;
    const size_t row = (size_t)blockIdx.x * 4 + wave;
    const float* h = H + row * (size_t)K;

    float acc[7];
#pragma unroll
    for (int j = 0; j < 7; ++j) acc[j] = 0.f;

    for (int k = lane; k < K; k += 32) {
        const float hv = h[k];
        const float* w = W2 + (size_t)k * N2;
        for (int j = 0; j < N2; ++j) acc[j] += hv * w[j];
    }
    for (int j = 0; j < N2; ++j)
        for (int s = 16; s > 0; s >>= 1)
            acc[j] += __shfl_xor(acc[j], s, 32);

    if (lane == 0)
        for (int j = 0; j < N2; ++j)
            outS[row * N2 + j] = acc[j] + b2[j];
}

// ---------------------------------------------------------------------------
// Per-row epilogue: softmax, expected size, tanh residual, clip, argmax route.
// ---------------------------------------------------------------------------
__global__ __launch_bounds__(256) void finalize_k(
    const float* __restrict__ L, const float* __restrict__ O,
    const float* __restrict__ R, float* __restrict__ out,
    int* __restrict__ sizeIdx)
{
    const size_t r = (size_t)blockIdx.x * 256 + threadIdx.x;
    float lg[7];
    float mx = -3.402823466e38f;
    int   am = 0;
#pragma unroll
    for (int j = 0; j < 7; ++j) {
        lg[j] = L[r * 7 + j];
        if (lg[j] > mx) { mx = lg[j]; am = j; }
    }
    float p[7], s = 0.f;
#pragma unroll
    for (int j = 0; j < 7; ++j) { p[j] = __expf(lg[j] - mx); s += p[j]; }
    const float inv = 1.f / s;
    float expd = 0.f;
#pragma unroll
    for (int j = 0; j < 7; ++j) { p[j] *= inv; expd += p[j] * (j * (1.f / 6.f)); }

    const float resid = 0.35f * tanhf(R[r]);
    float reg = expd + resid;
    reg = reg < 0.f ? 0.f : (reg > 1.f ? 1.f : reg);

    float* o = out + r * 24;
#pragma unroll
    for (int j = 0; j < 7; ++j) o[j] = lg[j];
#pragma unroll
    for (int j = 0; j < 6; ++j) o[7 + j] = O[r * 6 + j];
    o[13] = reg;
#pragma unroll
    for (int j = 0; j < 7; ++j) o[14 + j] = p[j];
    sizeIdx[r] = am;
}

// ---------------------------------------------------------------------------
// Counting-sort routing: zero -> count -> scan -> scatter.
// ---------------------------------------------------------------------------
__global__ void route_zero(int* __restrict__ c)
{
    if (threadIdx.x < 24) c[threadIdx.x] = 0;
}
__global__ __launch_bounds__(256) void route_count(
    const int* __restrict__ sizeIdx, int* __restrict__ cnt)
{
    const size_t r = (size_t)blockIdx.x * 256 + threadIdx.x;
    atomicAdd(&cnt[sizeIdx[r]], 1);
}
__global__ void route_scan(const int* __restrict__ cnt, int* __restrict__ off)
{
    if (threadIdx.x == 0) {
        int s = 0;
        for (int e2 = 0; e2 < 7; ++e2) { off[e2] = s; s += cnt[e2]; }
    }
}
__global__ __launch_bounds__(256) void route_scatter(
    const int* __restrict__ sizeIdx, const int* __restrict__ off,
    int* __restrict__ cur, int* __restrict__ perm)
{
    const size_t r = (size_t)blockIdx.x * 256 + threadIdx.x;
    const int e2 = sizeIdx[r];
    const int p  = atomicAdd(&cur[e2], 1);
    perm[off[e2] + p] = (int)r;
}

// ---------------------------------------------------------------------------
// Final expert layer (N=3): wave-per-row GEMV with per-row expert weights.
// ---------------------------------------------------------------------------
__global__ __launch_bounds__(128) void depth_out_k(
    const float* __restrict__ H2, const int* __restrict__ sizeIdx,
    const float* __restrict__ We3, const float* __restrict__ be3,
    float* __restrict__ out)
{
    const int lane = threadIdx.x & 31;
    const int wave = threadIdx.x >> 5;
    const size_t row = (size_t)blockIdx.x * 4 + wave;
    const int e = sizeIdx[row];
    const float* h = H2 + row * (size_t)DHID;
    const float* w = We3 + (size_t)e * DHID * 3;

    float a0 = 0.f, a1 = 0.f, a2 = 0.f;
    for (int k = lane; k < DHID; k += 32) {
        const float hv = h[k];
        a0 += hv * w[k * 3 + 0];
        a1 += hv * w[k * 3 + 1];
        a2 += hv * w[k * 3 + 2];
    }
    for (int s = 16; s > 0; s >>= 1) {
        a0 += __shfl_xor(a0, s, 32);
        a1 += __shfl_xor(a1, s, 32);
        a2 += __shfl_xor(a2, s, 32);
    }
    if (lane == 0) {
        float* o = out + row * 24;
        o[21] = a0 + be3[e * 3 + 0];
        o[22] = a1 + be3[e * 3 + 1];
        o[23] = a2 + be3[e * 3 + 2];
    }
}

// ---------------------------------------------------------------------------
extern "C" void kernel_launch(void* const* d_in, const int* in_sizes, int n_in,
                              void* d_out, int out_size, void* d_ws, size_t ws_size,
                              hipStream_t stream)
{
    (void)in_sizes; (void)n_in; (void)out_size; (void)ws_size;

    const float* X   = (const float*)d_in[0];
    const float* Wc1 = (const float*)d_in[1];
    const float* bc1 = (const float*)d_in[2];
    const float* Wc2 = (const float*)d_in[3];
    const float* bc2 = (const float*)d_in[4];
    const float* Wo1 = (const float*)d_in[5];
    const float* bo1 = (const float*)d_in[6];
    const float* Wo2 = (const float*)d_in[7];
    const float* bo2 = (const float*)d_in[8];
    const float* Wr1 = (const float*)d_in[9];
    const float* br1 = (const float*)d_in[10];
    const float* Wr2 = (const float*)d_in[11];
    const float* br2 = (const float*)d_in[12];
    const float* Wa  = (const float*)d_in[13];
    const float* ba  = (const float*)d_in[14];
    const float* We1 = (const float*)d_in[15];
    const float* be1 = (const float*)d_in[16];
    const float* We2 = (const float*)d_in[17];
    const float* be2 = (const float*)d_in[18];
    const float* We3 = (const float*)d_in[19];
    const float* be3 = (const float*)d_in[20];
    float* out = (float*)d_out;

    // Workspace layout (peak ~135 MB)
    char* ws = (char*)d_ws;
    const size_t bigBytes = (size_t)B_ROWS * DHID * sizeof(float);
    float* bufA = (float*)ws;                         // [B,1024] H / depth_feat / h2
    float* bufB = (float*)(ws + bigBytes);            // [B,1024] h1
    float* bufL = (float*)(ws + 2 * bigBytes);        // [B,7] size logits
    float* bufO = bufL + (size_t)B_ROWS * 7;          // [B,6] ord logits
    float* bufR = bufO + (size_t)B_ROWS * 6;          // [B]   residual raw
    int*   sizeIdx = (int*)(bufR + B_ROWS);           // [B]
    int*   perm    = sizeIdx + B_ROWS;                // [B]
    int*   cnt     = perm + B_ROWS;                   // [8]
    int*   offA    = cnt + 8;                         // [8]
    int*   cur     = cnt + 16;                        // [8]

    const dim3 blk(128);
    const dim3 g1024(1024 / 64, B_ROWS / 16);
    const dim3 g512(512 / 64, B_ROWS / 16);
    const dim3 gRow(B_ROWS / 4);
    const dim3 gFlat(B_ROWS / 256);

    // size classification head
    gemm_relu_wmma<<<g1024, blk, 0, stream>>>(X, Wc1, bc1, bufA,
        nullptr, nullptr, nullptr, 0, 1024, 1024, 1);
    head2_gemv<<<gRow, blk, 0, stream>>>(bufA, Wc2, bc2, bufL, 7, 1024);

    // ordinal head
    gemm_relu_wmma<<<g512, blk, 0, stream>>>(X, Wo1, bo1, bufA,
        nullptr, nullptr, nullptr, 0, 512, 1024, 1);
    head2_gemv<<<gRow, blk, 0, stream>>>(bufA, Wo2, bo2, bufO, 6, 512);

    // residual head
    gemm_relu_wmma<<<g1024, blk, 0, stream>>>(X, Wr1, br1, bufA,
        nullptr, nullptr, nullptr, 0, 1024, 1024, 1);
    head2_gemv<<<gRow, blk, 0, stream>>>(bufA, Wr2, br2, bufR, 1, 1024);

    // depth adapter -> depth_feat in bufA
    gemm_relu_wmma<<<g1024, blk, 0, stream>>>(X, Wa, ba, bufA,
        nullptr, nullptr, nullptr, 0, 1024, 1024, 1);

    // softmax / tanh / clip / argmax epilogue, writes out[:, 0:21] + route idx
    finalize_k<<<gFlat, dim3(256), 0, stream>>>(bufL, bufO, bufR, out, sizeIdx);

    // counting-sort rows by expert
    route_zero<<<dim3(1), dim3(32), 0, stream>>>(cnt);
    route_count<<<gFlat, dim3(256), 0, stream>>>(sizeIdx, cnt);
    route_scan<<<dim3(1), dim3(32), 0, stream>>>(cnt, offA);
    route_scatter<<<gFlat, dim3(256), 0, stream>>>(sizeIdx, offA, cur, perm);

    // hard-routed expert layer 1: h1 = relu(depth_feat @ We1[e]) (gather/scatter)
    for (int e = 0; e < 7; ++e)
        gemm_relu_wmma<<<g1024, blk, 0, stream>>>(bufA,
            We1 + (size_t)e * DHID * DHID, be1 + (size_t)e * DHID,
            bufB, perm, offA, cnt, e, 1024, 1024, 1);

    // expert layer 2: h2 = relu(h1 @ We2[e])
    for (int e = 0; e < 7; ++e)
        gemm_relu_wmma<<<g1024, blk, 0, stream>>>(bufB,
            We2 + (size_t)e * DHID * DHID, be2 + (size_t)e * DHID,
            bufA, perm, offA, cnt, e, 1024, 1024, 1);

    // expert output layer (N=3) -> out[:, 21:24]
    depth_out_k<<<gRow, blk, 0, stream>>>(bufA, sizeIdx, We3, be3, out);
}